// DWT_20315195310494
// MI455X (gfx1250) — compile-verified
//
#include <hip/hip_runtime.h>

// ---------------------------------------------------------------------------
// 2-D DWT (coif1, symmetric) for x:(8,32,512,512) f32.
//   out = [ LL : 256*258*258 ][ 4-band tensor : 256*4*258*258 ]
// Strategy (memory-bound, ~610MB @ 23.3TB/s => ~26us floor):
//   phase 1: async global->LDS copy of reflect-padded 36x132 input window
//   phase 2: horizontal stride-2 6-tap pass (VALU), 2 intermediates in LDS
//   phase 3: vertical pass as banded GEMM on v_wmma_f32_16x16x4_f32
//   stores : non-temporal (streaming, don't thrash L2)
// ---------------------------------------------------------------------------

typedef float v2f __attribute__((ext_vector_type(2)));
typedef float v8f __attribute__((ext_vector_type(8)));

#define H_IN 512
#define W_IN 512
#define N_OUT 258                 // (512 + 6 - 1) / 2
#define SP (N_OUT * N_OUT)        // 66564 per-plane
#define TOY 16                    // output rows per WG  (one WMMA M tile)
#define TOX 64                    // output cols per WG  (4 waves x N=16)
#define RE 36                     // extended rows needed = 2*TOY + 4
#define CE 132                    // extended cols needed = 2*TOX + 4
#define SIN 133                   // LDS stride of input tile
#define SMID 66                   // LDS stride of intermediates

// reversed decomposition filters: fr[s] = DEC[5-s]
#define FRL0 -0.0727326195128539f
#define FRL1  0.3378976624578092f
#define FRL2  0.8525720202122554f
#define FRL3  0.38486484686420286f
#define FRL4 -0.0727326195128539f
#define FRL5 -0.01565572813546454f
#define FRH0 -0.01565572813546454f
#define FRH1  0.0727326195128539f
#define FRH2  0.38486484686420286f
#define FRH3 -0.8525720202122554f
#define FRH4  0.3378976624578092f
#define FRH5  0.0727326195128539f

typedef __attribute__((address_space(1))) int gint;   // global (AS1)
typedef __attribute__((address_space(3))) int lint;   // LDS (AS3)

#if __has_builtin(__builtin_amdgcn_global_load_async_to_lds_b32)
#define HAVE_ASYNC_LDS 1
#else
#define HAVE_ASYNC_LDS 0
#endif

__device__ __forceinline__ void async_copy_b32(const float* g, float* l) {
#if HAVE_ASYNC_LDS
  __builtin_amdgcn_global_load_async_to_lds_b32(
      (gint*)const_cast<float*>(g), (lint*)l, 0, 0);
#else
  *l = __builtin_nontemporal_load(g);
#endif
}

__device__ __forceinline__ void async_wait_all() {
#if HAVE_ASYNC_LDS
#if __has_builtin(__builtin_amdgcn_s_wait_asynccnt)
  __builtin_amdgcn_s_wait_asynccnt(0);
#else
  asm volatile("s_wait_asynccnt 0" ::: "memory");
#endif
#endif
}

// symmetric extension index for length-512 axis; input t = ext_index - 4,
// valid for t in [-4, 639]  ->  [-t-1 | t | 1023-t]
__device__ __forceinline__ int reflect512(int t) {
  t = (t < 0) ? (-1 - t) : t;
  t = (t > 511) ? (1023 - t) : t;
  return t;
}

// branchless 6-way tap select (d dynamic per lane)
__device__ __forceinline__ float sel6(int d, float f0, float f1, float f2,
                                      float f3, float f4, float f5) {
  float v = 0.0f;
  v = (d == 0) ? f0 : v;
  v = (d == 1) ? f1 : v;
  v = (d == 2) ? f2 : v;
  v = (d == 3) ? f3 : v;
  v = (d == 4) ? f4 : v;
  v = (d == 5) ? f5 : v;
  return v;
}

__device__ __forceinline__ v8f wmma_f32_k4(v2f a, v2f b, v8f c) {
  // D = A(16x4,f32) x B(4x16,f32) + C   -> v_wmma_f32_16x16x4_f32
  return __builtin_amdgcn_wmma_f32_16x16x4_f32(
      /*neg_a=*/false, a, /*neg_b=*/false, b,
      /*c_mod=*/(short)0, c, /*reuse_a=*/false, /*reuse_b=*/false);
}

__global__ __launch_bounds__(128) void dwt2_coif1_kernel(
    const float* __restrict__ in, float* __restrict__ outLL,
    float* __restrict__ outT) {
  __shared__ float s_in[RE * SIN];   // reflect-padded input window
  __shared__ float s_lo[RE * SMID];  // row-pass low-band intermediate
  __shared__ float s_hi[RE * SMID];  // row-pass high-band intermediate

  const int tid = threadIdx.x;
  const int img = blockIdx.z;        // b*32 + c
  const int oy0 = blockIdx.y * TOY;
  const int ox0 = blockIdx.x * TOX;

  const float* src = in + (size_t)img * (H_IN * W_IN);

  // ---- phase 1: stage input window via CDNA5 async global->LDS ------------
  for (int i = tid; i < RE * CE; i += 128) {
    const int er = i / CE;
    const int ec = i - er * CE;
    const int gr = reflect512(2 * oy0 + er - 4);
    const int gc = reflect512(2 * ox0 + ec - 4);
    async_copy_b32(src + (size_t)gr * W_IN + gc, &s_in[er * SIN + ec]);
  }
  async_wait_all();
  __syncthreads();

  // ---- phase 2: horizontal stride-2 6-tap pass ----------------------------
  {
    const int oc = tid & (TOX - 1);
    const int r0 = tid >> 6;  // 0 or 1
    for (int r = r0; r < RE; r += 2) {
      const float* row = &s_in[r * SIN + 2 * oc];
      const float e0 = row[0], e1 = row[1], e2 = row[2];
      const float e3 = row[3], e4 = row[4], e5 = row[5];
      float lo = FRL0 * e0;
      lo = fmaf(FRL1, e1, lo); lo = fmaf(FRL2, e2, lo);
      lo = fmaf(FRL3, e3, lo); lo = fmaf(FRL4, e4, lo);
      lo = fmaf(FRL5, e5, lo);
      float hi = FRH0 * e0;
      hi = fmaf(FRH1, e1, hi); hi = fmaf(FRH2, e2, hi);
      hi = fmaf(FRH3, e3, hi); hi = fmaf(FRH4, e4, hi);
      hi = fmaf(FRH5, e5, hi);
      s_lo[r * SMID + oc] = lo;
      s_hi[r * SMID + oc] = hi;
    }
  }
  __syncthreads();

  // ---- phase 3: vertical pass = banded GEMM on the WMMA pipes -------------
  // D(16x16) = F(16x36) * X(36x16);  F[m,k] = f[k-2m] (0 outside [0,6))
  // A layout (16x4 f32): lane m=lane&15, half h=lane>>4, elem v -> K=v+2h
  // C/D layout: elem r -> M=r+8h, N=lane&15
  const int lane = tid & 31;
  const int wv = tid >> 5;            // 0..3 : which 16-col slice
  const int h = (lane >> 4) & 1;
  const int m = lane & 15;
  const int cb = wv * 16;

  v2f Alo[9], Ahi[9];
#pragma unroll
  for (int q = 0; q < 9; ++q) {
#pragma unroll
    for (int v = 0; v < 2; ++v) {
      const int k = 4 * q + v + 2 * h;
      const int d = k - 2 * m;
      Alo[q][v] = sel6(d, FRL0, FRL1, FRL2, FRL3, FRL4, FRL5);
      Ahi[q][v] = sel6(d, FRH0, FRH1, FRH2, FRH3, FRH4, FRH5);
    }
  }

  v8f accA = {}, accHH = {}, accV = {}, accD = {};
#pragma unroll
  for (int q = 0; q < 9; ++q) {
    const int k = 4 * q + 2 * h;      // K rows v=0,1 of this chunk
    v2f bl, bh;
    bl.x = s_lo[(k + 0) * SMID + cb + m];
    bl.y = s_lo[(k + 1) * SMID + cb + m];
    bh.x = s_hi[(k + 0) * SMID + cb + m];
    bh.y = s_hi[(k + 1) * SMID + cb + m];
    accA  = wmma_f32_k4(Alo[q], bl, accA);   // LL : lo(W) lo(H)
    accHH = wmma_f32_k4(Ahi[q], bl, accHH);  // H  : lo(W) hi(H)
    accV  = wmma_f32_k4(Alo[q], bh, accV);   // V  : hi(W) lo(H)
    accD  = wmma_f32_k4(Ahi[q], bh, accD);   // D  : hi(W) hi(H)
  }

  // ---- streaming stores ---------------------------------------------------
  const int ox = ox0 + cb + m;
  const size_t llbase = (size_t)img * SP;
  const size_t tbase = (size_t)img * 4 * SP;
#pragma unroll
  for (int r = 0; r < 8; ++r) {
    const int oy = oy0 + r + 8 * h;
    if (oy < N_OUT && ox < N_OUT) {
      const size_t sp = (size_t)oy * N_OUT + ox;
      __builtin_nontemporal_store(accA[r], &outLL[llbase + sp]);
      __builtin_nontemporal_store(accA[r], &outT[tbase + 0 * SP + sp]);
      __builtin_nontemporal_store(accHH[r], &outT[tbase + 1 * SP + sp]);
      __builtin_nontemporal_store(accV[r], &outT[tbase + 2 * SP + sp]);
      __builtin_nontemporal_store(accD[r], &outT[tbase + 3 * SP + sp]);
    }
  }
}

extern "C" void kernel_launch(void* const* d_in, const int* in_sizes, int n_in,
                              void* d_out, int out_size, void* d_ws,
                              size_t ws_size, hipStream_t stream) {
  (void)in_sizes; (void)n_in; (void)out_size; (void)d_ws; (void)ws_size;
  const float* x = (const float*)d_in[0];
  float* outLL = (float*)d_out;
  float* outT = outLL + (size_t)256 * SP;   // LL block then 4-band tensor
  dim3 grid((N_OUT + TOX - 1) / TOX,        // 5
            (N_OUT + TOY - 1) / TOY,        // 17
            256);                           // B*C images
  dim3 block(128);
  hipLaunchKernelGGL(dwt2_coif1_kernel, grid, block, 0, stream, x, outLL, outT);
}